// MLPKernel_4303557231270
// MI455X (gfx1250) — compile-verified
//
#include <hip/hip_runtime.h>
#include <cstdint>

typedef float v2f __attribute__((ext_vector_type(2)));
typedef float v4f __attribute__((ext_vector_type(4)));
typedef float v8f __attribute__((ext_vector_type(8)));

#define HH 64
#define N2_ 1024

// One block per (m, n1) row: 1024 points, 256 threads = 8 wave32.
// Layer2 ([1024x64] x [64x64]) runs on V_WMMA_F32_16X16X4_F32 (full f32).
// Register budget plan: pin W2 (WMMA B layout) in 128 VGPRs for the whole
// kernel; stream the W1 pack from LDS. asm memory clobbers (per pass and
// every 4 k-steps) bound the LDS-load staging window so peak pressure stays
// under 256 VGPRs: no VGPR-MSB addressing, no spills, ~4 waves/SIMD.
__global__ __launch_bounds__(256) void mlp_softmax_kernel(
    const float* __restrict__ diff, const uint8_t* __restrict__ mask,
    const float* __restrict__ W1, const float* __restrict__ b1,
    const float* __restrict__ W2, const float* __restrict__ b2,
    const float* __restrict__ W3, const float* __restrict__ b3,
    float* __restrict__ out)
{
    __shared__ v4f   sW1P[64];      // {W1[0][k], W1[1][k], W1[2][k], b1[k]}
    __shared__ v2f   sW2B[2048];    // W2 pre-packed into WMMA B-operand layout
    __shared__ float sScores[1024]; // per-row logits
    __shared__ float sRed[8];
    __shared__ float sBcast;

    const int tid  = threadIdx.x;
    const int lane = tid & 31;
    const int wave = tid >> 5;
    const int half = lane >> 4;    // which 16-lane half
    const int lp   = lane & 15;
    const int row  = blockIdx.x;   // m*N1 + n1

    // ---- stage weights into LDS ----
    if (tid < 64)
        sW1P[tid] = (v4f){W1[0*HH + tid], W1[1*HH + tid], W1[2*HH + tid], b1[tid]};
    for (int idx = tid; idx < 2048; idx += 256) {
        // idx = ((kk*4 + nt)*32 + ln): B tile for k-step kk, n-tile nt.
        // B 4x16 f32 layout: lanes 0-15 hold K rows {0,1}, lanes 16-31 hold {2,3}.
        int kk = idx >> 7;
        int nt = (idx >> 5) & 3;
        int ln = idx & 31;
        int kb = 4*kk + 2*(ln >> 4);
        int n  = 16*nt + (ln & 15);
        sW2B[idx] = (v2f){W2[kb*HH + n], W2[(kb + 1)*HH + n]};
    }
    __syncthreads();

    // ---- pin all 64 B tiles (W2 in WMMA layout) in 128 registers ----
    v2f Breg[64];
#pragma unroll
    for (int i = 0; i < 64; ++i)
        Breg[i] = sW2B[i*32 + lane];       // ds_load_b64, conflict-free, once

    // per-lane broadcast constants (lane lp owns column 16*nt+lp of b2/W3)
    float b2r[4], w3r[4];
#pragma unroll
    for (int nt = 0; nt < 4; ++nt) {
        b2r[nt] = b2[16*nt + lp];
        w3r[nt] = W3[16*nt + lp];
    }
    const float b3v = b3[0];

    // ---- 8 chunks of 16 points per wave ----
    for (int pass = 0; pass < 8; ++pass) {
        // Block LICM from hoisting this pass's LDS (sW1P) loads out of the loop.
        asm volatile("" ::: "memory");

        const int    chunk = pass*8 + wave;
        const int    n2    = chunk*16 + lp;                 // this lane's point (A-row M)
        const size_t pb    = ((size_t)row * N2_ + n2) * 3;
        const float  d0 = diff[pb], d1 = diff[pb + 1], d2 = diff[pb + 2];

        v8f acc[4];
#pragma unroll
        for (int nt = 0; nt < 4; ++nt)
#pragma unroll
            for (int r = 0; r < 8; ++r) acc[nt][r] = b2r[nt];  // C seeded with b2

#pragma unroll
        for (int kk = 0; kk < 16; ++kk) {
            // Bound the sW1P staging window: at most 4 k-steps (8 v4f) in
            // flight, so peak VGPR pressure stays below 256.
            if (kk == 4 || kk == 8 || kk == 12)
                asm volatile("" ::: "memory");

            // layer1 on the fly, directly in A layout: lane half h supplies K=4kk+2h,+1
            const int k0 = 4*kk + 2*half;
            v4f wa = sW1P[k0];
            v4f wb = sW1P[k0 + 1];
            float ax = fmaxf(wa.w + d0*wa.x + d1*wa.y + d2*wa.z, 0.f);
            float ay = fmaxf(wb.w + d0*wb.x + d1*wb.y + d2*wb.z, 0.f);
            v2f A = (v2f){ax, ay};
#pragma unroll
            for (int nt = 0; nt < 4; ++nt) {
                acc[nt] = __builtin_amdgcn_wmma_f32_16x16x4_f32(
                    false, A, false, Breg[kk*4 + nt], (short)0, acc[nt], false, false);
            }
        }

        // layer3 + relu: D layout row M = r + 8*half lives in acc[*][r]
#pragma unroll
        for (int r = 0; r < 8; ++r) {
            float t = 0.f;
#pragma unroll
            for (int nt = 0; nt < 4; ++nt)
                t += fmaxf(acc[nt][r], 0.f) * w3r[nt];
            // sum the 16 lanes of each half (xor masks < 16 never cross halves)
#pragma unroll
            for (int off = 1; off < 16; off <<= 1)
                t += __shfl_xor(t, off, 32);
            if (lp == r) sScores[chunk*16 + r + 8*half] = t + b3v;
        }
    }
    __syncthreads();

    // ---- masked softmax over the 1024 logits ----
    float s[4];
    float lmax = -__builtin_inff();
#pragma unroll
    for (int i = 0; i < 4; ++i) {
        int j = tid + i*256;
        float v = sScores[j];
        if (mask[(size_t)row * N2_ + j]) v = -__builtin_inff();
        s[i] = v;
        lmax = fmaxf(lmax, v);
    }
#pragma unroll
    for (int off = 16; off >= 1; off >>= 1) lmax = fmaxf(lmax, __shfl_xor(lmax, off, 32));
    if (lane == 0) sRed[wave] = lmax;
    __syncthreads();
    if (wave == 0) {
        float v = sRed[lane & 7];
#pragma unroll
        for (int off = 4; off >= 1; off >>= 1) v = fmaxf(v, __shfl_xor(v, off, 32));
        if (lane == 0) sBcast = v;
    }
    __syncthreads();
    const float gmax = sBcast;

    float e[4];
    float lsum = 0.f;
#pragma unroll
    for (int i = 0; i < 4; ++i) { e[i] = __expf(s[i] - gmax); lsum += e[i]; }
#pragma unroll
    for (int off = 16; off >= 1; off >>= 1) lsum += __shfl_xor(lsum, off, 32);
    if (lane == 0) sRed[wave] = lsum;
    __syncthreads();
    if (wave == 0) {
        float v = sRed[lane & 7];
#pragma unroll
        for (int off = 4; off >= 1; off >>= 1) v += __shfl_xor(v, off, 32);
        if (lane == 0) sBcast = v;
    }
    __syncthreads();
    const float inv = 1.f / sBcast;
#pragma unroll
    for (int i = 0; i < 4; ++i)
        out[(size_t)row * N2_ + tid + i*256] = e[i] * inv;
}

extern "C" void kernel_launch(void* const* d_in, const int* in_sizes, int n_in,
                              void* d_out, int out_size, void* d_ws, size_t ws_size,
                              hipStream_t stream) {
    (void)in_sizes; (void)n_in; (void)out_size; (void)d_ws; (void)ws_size;
    const float*   diff = (const float*)d_in[0];
    const uint8_t* mask = (const uint8_t*)d_in[1];   // JAX bool = 1 byte/elem
    const float*   W1   = (const float*)d_in[2];
    const float*   b1   = (const float*)d_in[3];
    const float*   W2   = (const float*)d_in[4];
    const float*   b2   = (const float*)d_in[5];
    const float*   W3   = (const float*)d_in[6];
    const float*   b3   = (const float*)d_in[7];
    float* out = (float*)d_out;

    dim3 grid(4 * 1024);   // one block per (m, n1) row
    mlp_softmax_kernel<<<grid, 256, 0, stream>>>(diff, mask, W1, b1, W2, b2, W3, b3, out);
}